// DiffPoolEncoder_25099788878068
// MI455X (gfx1250) — compile-verified
//
#include <hip/hip_runtime.h>
#include <hip/hip_bf16.h>
#include <math.h>

typedef __attribute__((ext_vector_type(2))) float v2f;
typedef __attribute__((ext_vector_type(4))) float v4f;
typedef __attribute__((ext_vector_type(8))) float v8f;

#define NPG  32            // nodes per graph
#define KC   10            // clusters
#define DD   300           // feature dim
#define BG   384           // graphs
#define NTOT (BG * NPG)    // 12288 total nodes
#define BK   (BG * KC)     // 3840 rows/cols of block-diag output
#define DT   19            // ceil(300/16) N-tiles for X_out
#define BK4  (BK / 4)      // 960 float4 per A_bd row

// D = A(16x4) * B(4x16) + C, fp32. One workgroup (128 thr = 4 wave32) per graph.
__global__ __launch_bounds__(128)
void diffpool_kernel(const float* __restrict__ S,
                     const float* __restrict__ A,
                     const float* __restrict__ X,
                     float* __restrict__ Xcat,   // [BK, DD]
                     float* __restrict__ Abd)    // [BK, BK]
{
    __shared__ float Pt[16][NPG];        // softmax(S_g), TRANSPOSED: [cluster][node], rows 10..15 zero
    __shared__ float Ash[NPG][NPG + 1];  // diagonal adjacency block (padded stride)
    __shared__ float Tl[16][NPG];        // T = P^T A_g  (16 x 32)
    __shared__ float Ao[16][16];         // A_out = T P  (only [0..9][0..9] valid)

    const int g    = blockIdx.x;
    const int tid  = threadIdx.x;
    const int lane = tid & 31;
    const int wave = tid >> 5;
    const int m    = lane & 15;          // WMMA row (A) / col (B,C,D) index
    const int ksel = (lane >> 4) * 2;    // K sub-pair select for half-wave
    const int rbase = (lane >> 4) * 8;   // C/D row base for this half-wave

    // ---- softmax over the 32 rows of S_g (one thread per node) ----
    if (tid < NPG) {
        const float* srow = S + (size_t)(g * NPG + tid) * KC;
        float v[KC];
        float mx = -INFINITY;
        #pragma unroll
        for (int c = 0; c < KC; ++c) { v[c] = srow[c]; mx = fmaxf(mx, v[c]); }
        float sum = 0.f;
        #pragma unroll
        for (int c = 0; c < KC; ++c) { v[c] = expf(v[c] - mx); sum += v[c]; }
        float inv = 1.0f / sum;
        #pragma unroll
        for (int c = 0; c < KC; ++c) Pt[c][tid] = v[c] * inv;
        #pragma unroll
        for (int c = KC; c < 16; ++c) Pt[c][tid] = 0.f;
    }

    // ---- stage the 32x32 diagonal block of A into LDS ----
    for (int e = tid; e < NPG * NPG; e += 128) {
        int i = e >> 5, j = e & 31;
        Ash[i][j] = A[(size_t)(g * NPG + i) * NTOT + (size_t)(g * NPG + j)];
    }
    __syncthreads();

    // ---- build the A-operand (P^T tile, 16x4 per k-step) once; reused everywhere ----
    // A[m,k] = P[node k][cluster m] = Pt[m][k]; contiguous even-aligned pair -> ds_load_b64
    v2f a_op[8];
    #pragma unroll
    for (int kk = 0; kk < 8; ++kk)
        a_op[kk] = *(const __shared__ v2f*)&Pt[m][kk * 4 + ksel];

    // ---- X_out = P^T X_g : 19 tiles of 16 cols, K=32 in 8 WMMA steps ----
    for (int t = wave; t < DT; t += 4) {
        const int col  = t * 16 + m;
        // Clamp instead of mask: lane n's B data only feeds output column n,
        // and the clamped lanes' columns (>=300, tile 18 only) are never stored.
        const int colc = col < DD ? col : (DD - 1);
        const float* xb = X + (size_t)(g * NPG + ksel) * DD + colc;
        v2f b[8];
        #pragma unroll
        for (int kk = 0; kk < 8; ++kk) {   // 16 loads, immediate offsets, one clause
            b[kk][0] = xb[kk * 4 * DD];
            b[kk][1] = xb[kk * 4 * DD + DD];
        }
        v8f acc = {};
        #pragma unroll
        for (int kk = 0; kk < 8; ++kk)
            acc = __builtin_amdgcn_wmma_f32_16x16x4_f32(
                      false, a_op[kk], false, b[kk], (short)0, acc, false, false);
        // C/D layout: VGPR r -> row r (lanes 0-15) / row r+8 (lanes 16-31).
        // Lanes 0-15 own cluster rows 0-7 (all 8 regs); lanes 16-31 own rows 8-9
        // (regs 0-1; regs 2-7 are pad rows 10-15, dropped). Two uniform-shaped
        // divergent regions -> two exec setups + store clauses, no per-store branch.
        const bool incol = (col < DD);
        if (incol && lane < 16) {
            float* o = Xcat + (size_t)(g * KC) * DD + col;
            #pragma unroll
            for (int r = 0; r < 8; ++r) o[(size_t)r * DD] = acc[r];
        }
        if (incol && lane >= 16) {
            float* o = Xcat + (size_t)(g * KC + 8) * DD + col;
            o[0]  = acc[0];
            o[DD] = acc[1];
        }
    }

    // ---- A_out = (P^T A_g) P on wave 0 ----
    if (wave == 0) {
        // T = P^T A_g : two 16-col tiles over the 32 node columns
        #pragma unroll
        for (int t = 0; t < 2; ++t) {
            const int nb = t * 16;
            v8f acc = {};
            #pragma unroll
            for (int kk = 0; kk < 8; ++kk) {
                const int r0 = kk * 4 + ksel;
                v2f b;
                b[0] = Ash[r0 + 0][nb + m];
                b[1] = Ash[r0 + 1][nb + m];
                acc = __builtin_amdgcn_wmma_f32_16x16x4_f32(
                          false, a_op[kk], false, b, (short)0, acc, false, false);
            }
            #pragma unroll
            for (int r = 0; r < 8; ++r) Tl[rbase + r][nb + m] = acc[r];
        }
        // A_out = T P : one 16x16 tile (rows/cols >= 10 ignored; Pt pad rows are zero)
        v8f acc = {};
        #pragma unroll
        for (int kk = 0; kk < 8; ++kk) {
            const int r0 = kk * 4 + ksel;
            v2f a = *(const __shared__ v2f*)&Tl[m][r0];   // A[m,k] = T[cluster m][node k]
            v2f b = *(const __shared__ v2f*)&Pt[m][r0];   // B[k,n] = P[node k][cluster n=m]
            acc = __builtin_amdgcn_wmma_f32_16x16x4_f32(
                      false, a, false, b, (short)0, acc, false, false);
        }
        #pragma unroll
        for (int r = 0; r < 8; ++r) Ao[rbase + r][m] = acc[r];
    }
    __syncthreads();

    // ---- write this graph's 10 full rows of A_bd: zeros + inline diagonal block ----
    float* rowbase = Abd + (size_t)g * KC * BK;
    const int g0 = g * KC;                  // block column start
    const int q0 = g0 >> 2;                 // first overlapping float4
    const int q1 = (g0 + KC - 1) >> 2;      // last overlapping float4
    for (int e = tid; e < KC * BK4; e += 128) {
        const int c1 = e / BK4;             // cluster row 0..9
        const int q  = e - c1 * BK4;        // float4 index within row
        v4f val = {0.f, 0.f, 0.f, 0.f};
        if (q >= q0 && q <= q1) {
            #pragma unroll
            for (int u = 0; u < 4; ++u) {
                const int c = q * 4 + u - g0;
                if (c >= 0 && c < KC) val[u] = Ao[c1][c];
            }
        }
        ((v4f*)rowbase)[(size_t)c1 * BK4 + q] = val;
    }
}

extern "C" void kernel_launch(void* const* d_in, const int* in_sizes, int n_in,
                              void* d_out, int out_size, void* d_ws, size_t ws_size,
                              hipStream_t stream) {
    (void)in_sizes; (void)n_in; (void)d_ws; (void)ws_size; (void)out_size;
    const float* S = (const float*)d_in[0];   // [N, K] fp32
    const float* A = (const float*)d_in[1];   // [N, N] fp32
    const float* X = (const float*)d_in[2];   // [N, D] fp32
    // d_in[3] = idx (unused: segment structure implied by n=32), d_in[4] = n (=32)
    float* Xcat = (float*)d_out;                          // [BK, DD]
    float* Abd  = (float*)d_out + (size_t)BK * DD;        // [BK, BK]
    diffpool_kernel<<<dim3(BG), dim3(128), 0, stream>>>(S, A, X, Xcat, Abd);
}